// GAT_48095043780693
// MI455X (gfx1250) — compile-verified
//
#include <hip/hip_runtime.h>
#include <math.h>

typedef __attribute__((ext_vector_type(2))) float v2f;
typedef __attribute__((ext_vector_type(8))) float v8f;

#define NEG_SLOPE 0.2f
#define ORDERED_NEG_INF ((int)0x807FFFFF)  // f2oi(-inf)

// ---- monotonic float<->int mapping for atomic float max ----
__device__ __forceinline__ int f2oi(float f) {
  int i = __float_as_int(f);
  return i >= 0 ? i : (i ^ 0x7FFFFFFF);
}
__device__ __forceinline__ float oi2f(int i) {
  return __int_as_float(i >= 0 ? i : (i ^ 0x7FFFFFFF));
}

__device__ __forceinline__ void edge_sd(const int* __restrict__ ei, int E, int eid,
                                        int& s, int& d) {
  if (eid < E) { s = ei[eid]; d = ei[E + eid]; }
  else { s = eid - E; d = s; }  // self loop
}

// ===================== fp32 WMMA GEMM: C[M,NCOLS] = A[M,K] @ B[K,NCOLS] =====
// One wave computes a 16-row tile across all column tiles using
// V_WMMA_F32_16X16X4_F32 (exact fp32 math, RNE). M must be a multiple of 16.
template<int K, int NCOLS>
__global__ void k_gemm_wmma(const float* __restrict__ A, const float* __restrict__ B,
                            float* __restrict__ C, int M) {
  constexpr int NT = (NCOLS + 15) / 16;
  int wid  = (int)((blockIdx.x * blockDim.x + threadIdx.x) >> 5);
  int lane = (int)(threadIdx.x & 31);
  int nRowTiles = (M + 15) >> 4;
  if (wid >= nRowTiles) return;          // whole-wave uniform exit: EXEC all-1 for WMMA
  int hs  = lane >> 4;                   // half-wave select (K pair 0/1 vs 2/3)
  int l15 = lane & 15;
  int row = wid * 16 + l15;

  v8f acc[NT];
#pragma unroll
  for (int t = 0; t < NT; ++t)
#pragma unroll
    for (int r = 0; r < 8; ++r) acc[t][r] = 0.0f;

  const float* arow = A + (long)row * K;
  for (int k0 = 0; k0 < K; k0 += 4) {
    int kb = k0 + 2 * hs;
    v2f a;                               // A 16x4 frag: lane<16 holds K=k0..k0+1, lane>=16 K=k0+2..k0+3
    a.x = arow[kb];
    a.y = arow[kb + 1];
#pragma unroll
    for (int t = 0; t < NT; ++t) {
      int col = t * 16 + l15;
      v2f b;                             // B 4x16 frag, symmetric layout
      if (col < NCOLS) {
        b.x = B[kb * NCOLS + col];
        b.y = B[(kb + 1) * NCOLS + col];
      } else { b.x = 0.0f; b.y = 0.0f; }
      acc[t] = __builtin_amdgcn_wmma_f32_16x16x4_f32(
          false, a, false, b, (short)0, acc[t], false, false);
    }
  }
#pragma unroll
  for (int t = 0; t < NT; ++t) {
    int col = t * 16 + l15;
    if (col < NCOLS) {
#pragma unroll
      for (int r = 0; r < 8; ++r) {
        int orow = wid * 16 + r + 8 * hs;   // C/D layout: vgpr r -> M=r (+8 for upper half-wave)
        C[(long)orow * NCOLS + col] = acc[t][r];
      }
    }
  }
}

// ===================== per-node attention logits ============================
template<int H, int C>
__global__ void k_alpha(const float* __restrict__ h, const float* __restrict__ a_src,
                        const float* __restrict__ a_dst, float* __restrict__ aS,
                        float* __restrict__ aD, int N) {
  long idx = (long)blockIdx.x * blockDim.x + threadIdx.x;
  if (idx >= (long)N * H) return;
  int n = (int)(idx / H), hh = (int)(idx % H);
  const float* hp = h + ((long)n * H + hh) * C;
  const float* sp = a_src + hh * C;
  const float* dp = a_dst + hh * C;
  float s = 0.0f, d = 0.0f;
#pragma unroll
  for (int c = 0; c < C; ++c) { float v = hp[c]; s += v * sp[c]; d += v * dp[c]; }
  aS[idx] = s; aD[idx] = d;
}

// ===================== edge passes ==========================================
template<int H>
__global__ void k_edge_max(const int* __restrict__ ei, int E, int ET,
                           const float* __restrict__ aS, const float* __restrict__ aD,
                           int* __restrict__ emax) {
  long idx = (long)blockIdx.x * blockDim.x + threadIdx.x;
  if (idx >= (long)ET * H) return;
  int eid = (int)(idx / H), hh = (int)(idx % H);
  int s, d; edge_sd(ei, E, eid, s, d);
  float e = aS[(long)s * H + hh] + aD[(long)d * H + hh];
  e = e > 0.0f ? e : NEG_SLOPE * e;
  atomicMax(&emax[(long)d * H + hh], f2oi(e));
}

template<int H>
__global__ void k_edge_expsum(const int* __restrict__ ei, int E, int ET,
                              const float* __restrict__ aS, const float* __restrict__ aD,
                              const int* __restrict__ emax, float* __restrict__ denom) {
  long idx = (long)blockIdx.x * blockDim.x + threadIdx.x;
  if (idx >= (long)ET * H) return;
  int eid = (int)(idx / H), hh = (int)(idx % H);
  int s, d; edge_sd(ei, E, eid, s, d);
  long ih = (long)d * H + hh;
  float e = aS[(long)s * H + hh] + aD[ih];
  e = e > 0.0f ? e : NEG_SLOPE * e;
  atomicAdd(&denom[ih], expf(e - oi2f(emax[ih])));
}

template<int H, int C>
__global__ void k_edge_agg(const int* __restrict__ ei, int E, int ET,
                           const float* __restrict__ aS, const float* __restrict__ aD,
                           const int* __restrict__ emax, const float* __restrict__ denom,
                           const float* __restrict__ hf, float* __restrict__ agg) {
  long idx = (long)blockIdx.x * blockDim.x + threadIdx.x;
  if (idx >= (long)ET * H * C) return;
  int eid = (int)(idx / (H * C));
  int r   = (int)(idx % (H * C));
  int hh  = r / C;
  int s, d; edge_sd(ei, E, eid, s, d);
  long ih = (long)d * H + hh;
  float e = aS[(long)s * H + hh] + aD[ih];
  e = e > 0.0f ? e : NEG_SLOPE * e;
  float w = expf(e - oi2f(emax[ih])) / (denom[ih] + 1e-16f);
  atomicAdd(&agg[(long)d * H * C + r], hf[(long)s * H * C + r] * w);
}

// ===================== misc elementwise =====================================
__global__ void k_fill_i32(int* __restrict__ p, long n, int v) {
  long i = (long)blockIdx.x * blockDim.x + threadIdx.x;
  if (i < n) p[i] = v;
}

__global__ void k_bias_relu(float* __restrict__ x, const float* __restrict__ b,
                            long n, int D) {
  long i = (long)blockIdx.x * blockDim.x + threadIdx.x;
  if (i >= n) return;
  float v = x[i] + b[(int)(i % D)];
  x[i] = v > 0.0f ? v : 0.0f;
}

__global__ void k_logsoftmax_bias(float* __restrict__ out, const float* __restrict__ b,
                                  int N, int D) {
  long n = (long)blockIdx.x * blockDim.x + threadIdx.x;
  if (n >= N) return;
  float* row = out + n * D;
  float m = -INFINITY;
  for (int c = 0; c < D; ++c) m = fmaxf(m, row[c] + b[c]);
  float sum = 0.0f;
  for (int c = 0; c < D; ++c) sum += expf(row[c] + b[c] - m);
  float lse = logf(sum);
  for (int c = 0; c < D; ++c) row[c] = row[c] + b[c] - m - lse;
}

// ===================== launcher =============================================
static inline int cdiv(long a, long b) { return (int)((a + b - 1) / b); }

extern "C" void kernel_launch(void* const* d_in, const int* in_sizes, int n_in,
                              void* d_out, int out_size, void* d_ws, size_t ws_size,
                              hipStream_t stream) {
  const float* x   = (const float*)d_in[0];
  const int*   ei  = (const int*)d_in[1];
  const float* W1  = (const float*)d_in[2];
  const float* as1 = (const float*)d_in[3];
  const float* ad1 = (const float*)d_in[4];
  const float* b1  = (const float*)d_in[5];
  const float* W2  = (const float*)d_in[6];
  const float* as2 = (const float*)d_in[7];
  const float* ad2 = (const float*)d_in[8];
  const float* b2  = (const float*)d_in[9];
  float* out = (float*)d_out;

  const int F = 128;
  int N  = in_sizes[0] / F;     // 100000
  int E  = in_sizes[1] / 2;     // 3200000
  int ET = E + N;               // + self loops

  // workspace layout (floats): h1/g2 | agg1/h2 | aS | aD | emax | denom
  float* h1    = (float*)d_ws;                     // N*64 (reused as g2)
  float* agg1  = h1   + (size_t)N * 64;            // N*64 (becomes h2)
  float* aS    = agg1 + (size_t)N * 64;            // N*8
  float* aD    = aS   + (size_t)N * 8;             // N*8
  int*   emax  = (int*)(aD + (size_t)N * 8);       // N*8
  float* denom = (float*)(emax + (size_t)N * 8);   // N*8

  const int B = 256;
  int gemmBlocks1 = cdiv((long)cdiv(N, 16) * 32, B);

  // -------- layer 1 --------
  k_gemm_wmma<128, 64><<<gemmBlocks1, B, 0, stream>>>(x, W1, h1, N);
  k_alpha<8, 8><<<cdiv((long)N * 8, B), B, 0, stream>>>(h1, as1, ad1, aS, aD, N);
  k_fill_i32<<<cdiv((long)N * 8, B), B, 0, stream>>>(emax, (long)N * 8, ORDERED_NEG_INF);
  hipMemsetAsync(denom, 0, (size_t)N * 8 * sizeof(float), stream);
  hipMemsetAsync(agg1, 0, (size_t)N * 64 * sizeof(float), stream);
  k_edge_max<8><<<cdiv((long)ET * 8, B), B, 0, stream>>>(ei, E, ET, aS, aD, emax);
  k_edge_expsum<8><<<cdiv((long)ET * 8, B), B, 0, stream>>>(ei, E, ET, aS, aD, emax, denom);
  k_edge_agg<8, 8><<<cdiv((long)ET * 64, B), B, 0, stream>>>(ei, E, ET, aS, aD, emax,
                                                             denom, h1, agg1);
  k_bias_relu<<<cdiv((long)N * 64, B), B, 0, stream>>>(agg1, b1, (long)N * 64, 64);

  // -------- layer 2 (g2 = h2 @ W2 into h1 buffer) --------
  k_gemm_wmma<64, 40><<<gemmBlocks1, B, 0, stream>>>(agg1, W2, h1, N);
  k_alpha<1, 40><<<cdiv((long)N, B), B, 0, stream>>>(h1, as2, ad2, aS, aD, N);
  k_fill_i32<<<cdiv((long)N, B), B, 0, stream>>>(emax, (long)N, ORDERED_NEG_INF);
  hipMemsetAsync(denom, 0, (size_t)N * sizeof(float), stream);
  hipMemsetAsync(out, 0, (size_t)N * 40 * sizeof(float), stream);
  k_edge_max<1><<<cdiv((long)ET, B), B, 0, stream>>>(ei, E, ET, aS, aD, emax);
  k_edge_expsum<1><<<cdiv((long)ET, B), B, 0, stream>>>(ei, E, ET, aS, aD, emax, denom);
  k_edge_agg<1, 40><<<cdiv((long)ET * 40, B), B, 0, stream>>>(ei, E, ET, aS, aD, emax,
                                                              denom, h1, out);
  k_logsoftmax_bias<<<cdiv((long)N, B), B, 0, stream>>>(out, b2, N, 40);
}